// Tree_Net_88923002896401
// MI455X (gfx1250) — compile-verified
//
#include <hip/hip_runtime.h>
#include <math.h>

// ---- problem constants (match reference) ----
#define BATCH      256
#define SEQ        64
#define EMB        1024
#define HID        512
#define MAX_NODE   127          // 2*SEQ-1
#define NUM_STEPS  63           // SEQ-1
#define NCAT       511
#define EPS        1e-6f

typedef __attribute__((ext_vector_type(2))) float v2f;
typedef __attribute__((ext_vector_type(8))) float v8f;

// ---- optional CDNA5 async global->LDS staging (ASYNCcnt path) ----
#if defined(__has_builtin)
#if __has_builtin(__builtin_amdgcn_global_load_async_to_lds_b128) && \
    __has_builtin(__builtin_amdgcn_s_wait_asynccnt)
#define USE_ASYNC_LDS 1
#endif
#endif

#if USE_ASYNC_LDS
typedef int v4i __attribute__((vector_size(16)));
typedef __attribute__((address_space(1))) v4i* gptr4;   // global int4*
typedef __attribute__((address_space(3))) v4i* lptr4;   // LDS int4*

__device__ __forceinline__ void async_cp16(const void* g, void* l) {
    __builtin_amdgcn_global_load_async_to_lds_b128(
        (gptr4)g, (lptr4)l, /*offset=*/0, /*cpol=*/0);
}
__device__ __forceinline__ void async_wait0() {
    __builtin_amdgcn_s_wait_asynccnt(0);
}
#endif

// D = A(16x4 f32) * B(4x16 f32) + C  via CDNA5 WMMA
// A frag: lanes 0-15 hold rows M=lane, {vgpr0,vgpr1}={K0,K1}; lanes16-31: {K2,K3}
// B frag (dual):      lanes 0-15 hold cols N=lane, {K0,K1}; lanes16-31: {K2,K3}
// C/D:  vgpr r, lane<16 -> (M=r, N=lane); lane>=16 -> (M=8+r, N=lane-16)
__device__ __forceinline__ v8f wmma4(v2f a, v2f b, v8f c) {
    return __builtin_amdgcn_wmma_f32_16x16x4_f32(
        /*neg_a=*/false, a, /*neg_b=*/false, b,
        /*c_mod=*/(short)0, c, /*reuse_a=*/false, /*reuse_b=*/false);
}

// =====================================================================
// Kernel 1: combined = relu( X[:, :512]@W1 + X[:, 512:]@W2 )
//   X = seq_rep viewed as (16384 x 1024), Wcat = [W1;W2] (1024 x 512)
//   128x64 block tile, BK=32, 8 waves stacked in M; each wave: 16(M)x64(N)
//   -> one A fragment feeds 4 WMMAs (better ds:wmma ratio for K=4 fp32)
// =====================================================================
__global__ void __launch_bounds__(256)
k_gemm_combine(const float* __restrict__ X, const float* __restrict__ W1,
               const float* __restrict__ W2, float* __restrict__ combined)
{
    __shared__ float As[128 * 36];  // ld 36 (pad) -> conflict-free frag reads
    __shared__ float Bs[32 * 68];   // ld 68 (pad, mult of 4 for float4 stores)

    const int mBase = blockIdx.x * 128;
    const int nBase = blockIdx.y * 64;
    const int tid   = threadIdx.x;
    const int lane  = tid & 31;
    const int wave  = tid >> 5;          // 0..7
    const int mOff  = wave * 16;
    const int fr    = lane & 15;
    const int kh    = (lane >> 4) << 1;  // 0 or 2

    v8f acc[4];
    #pragma unroll
    for (int nt = 0; nt < 4; ++nt)
        acc[nt] = (v8f){0.f,0.f,0.f,0.f,0.f,0.f,0.f,0.f};

    for (int kt = 0; kt < EMB; kt += 32) {
        // ---- stage A tile: rows mBase..+128, cols kt..+32 (1024 float4) ----
        #pragma unroll
        for (int it = 0; it < 4; ++it) {
            int v = tid + it * 256;            // 0..1023
            int row = v >> 3, col4 = (v & 7) * 4;
            const float* gp = X + (size_t)(mBase + row) * EMB + kt + col4;
#if USE_ASYNC_LDS
            async_cp16(gp, &As[row * 36 + col4]);
#else
            *(float4*)(&As[row * 36 + col4]) = *(const float4*)gp;
#endif
        }
        // ---- stage B tile: Wcat rows kt..+32, cols nBase..+64 ----
        #pragma unroll
        for (int it = 0; it < 2; ++it) {
            int v = tid + it * 256;            // 0..511
            int row = v >> 4, col4 = (v & 15) * 4;
            int k = kt + row;
            const float* wsrc = ((k < HID) ? (W1 + (size_t)k * HID)
                                           : (W2 + (size_t)(k - HID) * HID))
                                + nBase + col4;
#if USE_ASYNC_LDS
            async_cp16(wsrc, &Bs[row * 68 + col4]);
#else
            *(float4*)(&Bs[row * 68 + col4]) = *(const float4*)wsrc;
#endif
        }
        // prefetch next A K-tile into L2 while we compute
        if (kt + 32 < EMB)
            __builtin_prefetch(X + (size_t)(mBase + (tid & 127)) * EMB + kt + 32, 0, 0);
#if USE_ASYNC_LDS
        async_wait0();
#endif
        __syncthreads();

        #pragma unroll
        for (int kk = 0; kk < 32; kk += 4) {
            v2f af; af.x = As[(mOff + fr) * 36 + kk + kh];
                    af.y = As[(mOff + fr) * 36 + kk + kh + 1];
            #pragma unroll
            for (int nt = 0; nt < 4; ++nt) {
                v2f bf; bf.x = Bs[(kk + kh) * 68 + nt * 16 + fr];
                        bf.y = Bs[(kk + kh + 1) * 68 + nt * 16 + fr];
                acc[nt] = wmma4(af, bf, acc[nt]);
            }
        }
        __syncthreads();
    }

    // ---- epilogue: relu + store ----
    const int rowB = mBase + mOff + ((lane >> 4) << 3);
    #pragma unroll
    for (int nt = 0; nt < 4; ++nt) {
        const int col = nBase + nt * 16 + fr;
        #pragma unroll
        for (int r = 0; r < 8; ++r) {
            float v = acc[nt][r]; v = v > 0.f ? v : 0.f;
            combined[(size_t)(rowB + r) * HID + col] = v;
        }
    }
}

// =====================================================================
// Kernel 2a: zero the node-vector buffer (ws is poisoned by harness)
// =====================================================================
__global__ void k_zero(float4* __restrict__ p, int n4)
{
    int i = blockIdx.x * blockDim.x + threadIdx.x;
    float4 z = make_float4(0.f, 0.f, 0.f, 0.f);
    for (; i < n4; i += gridDim.x * blockDim.x) p[i] = z;
}

// =====================================================================
// Kernel 2b: vector[b, pos0] = normalize(combined[b, pos1])
//   one 128-thread block per (b,s); 4 floats per thread
// =====================================================================
__global__ void __launch_bounds__(128)
k_scatter_norm(const float* __restrict__ combined, const int* __restrict__ opos,
               float* __restrict__ vec)
{
    const int b = blockIdx.x >> 6;
    const int s = blockIdx.x & 63;
    const int p0 = opos[((size_t)b * SEQ + s) * 2 + 0];
    const int p1 = opos[((size_t)b * SEQ + s) * 2 + 1];
    const int tid = threadIdx.x;

    const float4 v = ((const float4*)(combined + ((size_t)b * SEQ + p1) * HID))[tid];
    float ss = v.x * v.x + v.y * v.y + v.z * v.z + v.w * v.w;

    __shared__ float red[128];
    red[tid] = ss;
    __syncthreads();
    for (int o = 64; o > 0; o >>= 1) {
        if (tid < o) red[tid] += red[tid + o];
        __syncthreads();
    }
    const float inv = 1.0f / (sqrtf(red[0]) + EPS);
    float4 o; o.x = v.x * inv; o.y = v.y * inv; o.z = v.z * inv; o.w = v.w * inv;
    ((float4*)(vec + ((size_t)b * MAX_NODE + p0) * HID))[tid] = o;
}

// =====================================================================
// Kernel 3: one tree composition step (launched 63x, sequential on stream)
//   Circular correlation as a per-batch GEMM on the matrix cores:
//     c[16q+n] = sum_u a[(u-16q)&511] * b[(u+n)&511]
//   block = 64 threads (2 waves); wave w owns q in [16w, 16w+16)
// =====================================================================
__global__ void __launch_bounds__(64)
k_tree_step(const int* __restrict__ cinfo, float* __restrict__ vec, int step)
{
    __shared__ float aL[HID];
    __shared__ float bR[HID];
    __shared__ float red[2];

    const int b = blockIdx.x;
    const int* info = cinfo + ((size_t)b * NUM_STEPS + step) * 4;
    const int t = info[0], p = info[1], l = info[2], r = info[3];
    float* base = vec + (size_t)b * MAX_NODE * HID;

    const int tid = threadIdx.x;              // 0..63
    const float4* ls = (const float4*)(base + (size_t)l * HID);
    const float4* rs = (const float4*)(base + (size_t)r * HID);
    #pragma unroll
    for (int i = 0; i < 2; ++i) {
#if USE_ASYNC_LDS
        async_cp16(ls + tid + i * 64, &((float4*)aL)[tid + i * 64]);
        async_cp16(rs + tid + i * 64, &((float4*)bR)[tid + i * 64]);
#else
        ((float4*)aL)[tid + i * 64] = ls[tid + i * 64];
        ((float4*)bR)[tid + i * 64] = rs[tid + i * 64];
#endif
    }
#if USE_ASYNC_LDS
    async_wait0();
#endif
    __syncthreads();

    const int wave = tid >> 5, lane = tid & 31;
    const int qBase = wave * 16;
    const int fr = lane & 15;
    const int kh = (lane >> 4) << 1;
    const int q  = qBase + fr;

    v8f acc = {0.f,0.f,0.f,0.f,0.f,0.f,0.f,0.f};
    for (int kb = 0; kb < HID; kb += 4) {
        const int u0 = kb + kh;
        v2f af; af.x = aL[(u0     - 16 * q) & (HID - 1)];
                af.y = aL[(u0 + 1 - 16 * q) & (HID - 1)];
        v2f bf; bf.x = bR[(u0     + fr) & (HID - 1)];
                bf.y = bR[(u0 + 1 + fr) & (HID - 1)];
        acc = wmma4(af, bf, acc);
    }

    // ---- ||c||: each lane holds 8 outputs; reduce wave then block ----
    float ss = 0.f;
    #pragma unroll
    for (int i = 0; i < 8; ++i) ss += acc[i] * acc[i];
    for (int m = 16; m > 0; m >>= 1) ss += __shfl_xor(ss, m, 32);
    if (lane == 0) red[wave] = ss;
    __syncthreads();
    const float inv = 1.0f / (sqrtf(red[0] + red[1]) + EPS);

    float* dst = base + (size_t)p * HID;
    if (t == 2) {
        const int mB = qBase + ((lane >> 4) << 3);
        #pragma unroll
        for (int rr = 0; rr < 8; ++rr)
            dst[(mB + rr) * 16 + fr] = acc[rr] * inv;   // c index = 16q + n
    } else if (t == 1) {
        #pragma unroll
        for (int i = 0; i < 8; ++i) dst[tid + i * 64] = aL[tid + i * 64];
    } // else: keep current value (matches jnp.where fall-through)
}

// =====================================================================
// Kernel 4: out = vec(32512x512) @ linear_w^T (512x511) + bias
//   128x64 tile like kernel 1; B staged transposed with N-edge guard
// =====================================================================
__global__ void __launch_bounds__(256)
k_gemm_out(const float* __restrict__ vec, const float* __restrict__ lw,
           const float* __restrict__ lb, float* __restrict__ out)
{
    __shared__ float As[128 * 36];
    __shared__ float Bs[32 * 68];

    const int mBase = blockIdx.x * 128;
    const int nBase = blockIdx.y * 64;
    const int tid   = threadIdx.x;
    const int lane  = tid & 31;
    const int wave  = tid >> 5;
    const int mOff  = wave * 16;
    const int fr    = lane & 15;
    const int kh    = (lane >> 4) << 1;

    v8f acc[4];
    #pragma unroll
    for (int nt = 0; nt < 4; ++nt)
        acc[nt] = (v8f){0.f,0.f,0.f,0.f,0.f,0.f,0.f,0.f};

    for (int kt = 0; kt < HID; kt += 32) {
        #pragma unroll
        for (int it = 0; it < 4; ++it) {
            int v = tid + it * 256;
            int row = v >> 3, col4 = (v & 7) * 4;
            const float* gp = vec + (size_t)(mBase + row) * HID + kt + col4;
#if USE_ASYNC_LDS
            async_cp16(gp, &As[row * 36 + col4]);
#else
            *(float4*)(&As[row * 36 + col4]) = *(const float4*)gp;
#endif
        }
        // B = lw^T : Bs[k][c] = lw[c, kt+k]; guard c >= NCAT with zeros
        {
            int c  = tid >> 2;                 // 0..63
            int k0 = (tid & 3) * 8;            // 0,8,16,24
            int cG = nBase + c;
            #pragma unroll
            for (int j = 0; j < 8; ++j) {
                int k = k0 + j;
                float val = (cG < NCAT) ? lw[(size_t)cG * HID + kt + k] : 0.f;
                Bs[k * 68 + c] = val;
            }
        }
        if (kt + 32 < HID)
            __builtin_prefetch(vec + (size_t)(mBase + (tid & 127)) * HID + kt + 32, 0, 0);
#if USE_ASYNC_LDS
        async_wait0();
#endif
        __syncthreads();

        #pragma unroll
        for (int kk = 0; kk < 32; kk += 4) {
            v2f af; af.x = As[(mOff + fr) * 36 + kk + kh];
                    af.y = As[(mOff + fr) * 36 + kk + kh + 1];
            #pragma unroll
            for (int nt = 0; nt < 4; ++nt) {
                v2f bf; bf.x = Bs[(kk + kh) * 68 + nt * 16 + fr];
                        bf.y = Bs[(kk + kh + 1) * 68 + nt * 16 + fr];
                acc[nt] = wmma4(af, bf, acc[nt]);
            }
        }
        __syncthreads();
    }

    const int rowB = mBase + mOff + ((lane >> 4) << 3);
    #pragma unroll
    for (int nt = 0; nt < 4; ++nt) {
        const int col = nBase + nt * 16 + fr;
        if (col < NCAT) {
            const float bias = lb[col];
            #pragma unroll
            for (int r = 0; r < 8; ++r)
                out[(size_t)(rowB + r) * NCAT + col] = acc[nt][r] + bias;
        }
    }
}

// =====================================================================
// Host launcher
// =====================================================================
extern "C" void kernel_launch(void* const* d_in, const int* in_sizes, int n_in,
                              void* d_out, int out_size, void* d_ws, size_t ws_size,
                              hipStream_t stream)
{
    const float* seq   = (const float*)d_in[0];   // (B,S,E)
    const float* W1    = (const float*)d_in[1];   // (H,H)
    const float* W2    = (const float*)d_in[2];   // (H,H)
    const float* lw    = (const float*)d_in[3];   // (NCAT,H)
    const float* lb    = (const float*)d_in[4];   // (NCAT,)
    const int*   opos  = (const int*)d_in[7];     // (B,S,2)
    const int*   cinfo = (const int*)d_in[8];     // (B,NUM_STEPS,4)
    float* out = (float*)d_out;

    float* combined = (float*)d_ws;                                   // 16384*512 f32
    float* vec      = combined + (size_t)BATCH * SEQ * HID;           // 32512*512 f32

    // 1) input GEMM + relu
    k_gemm_combine<<<dim3((BATCH * SEQ) / 128, HID / 64), 256, 0, stream>>>(
        seq, W1, W2, combined);

    // 2) zero node buffer, scatter + row-normalize leaves
    {
        int n4 = (BATCH * MAX_NODE * HID) / 4;
        k_zero<<<2048, 256, 0, stream>>>((float4*)vec, n4);
    }
    k_scatter_norm<<<BATCH * SEQ, 128, 0, stream>>>(combined, opos, vec);

    // 3) 63 sequential tree composition steps (WMMA circulant GEMM per batch)
    for (int s = 0; s < NUM_STEPS; ++s)
        k_tree_step<<<BATCH, 64, 0, stream>>>(cinfo, vec, s);

    // 4) output GEMM + bias
    k_gemm_out<<<dim3((BATCH * MAX_NODE) / 128, 8), 256, 0, stream>>>(
        vec, lw, lb, out);
}